// String_Finder_Tree_38259568673536
// MI455X (gfx1250) — compile-verified
//
#include <hip/hip_runtime.h>
#include <math.h>

#define IMG   256
#define NPTS  8192
#define KNN   8
#define RAD2  144.0f
#define ROWS_PER_WAVE 32
#define COL_WAVES     8                    // waves per block, each scans 1/8 of columns
#define COLS_PER_WAVE (NPTS / COL_WAVES)   // 1024

typedef __attribute__((ext_vector_type(2))) float v2f;
typedef __attribute__((ext_vector_type(8))) float v8f;

// ---------------------------------------------------------------------------
// Kernel 1: Sobel gradient sampled at the 8192 points, normalized.
// ---------------------------------------------------------------------------
__global__ __launch_bounds__(256)
void sobel_norms_kernel(const float* __restrict__ batch,
                        const float* __restrict__ locs,
                        const int*   __restrict__ imgid,
                        float*       __restrict__ norms)
{
    int n = blockIdx.x * blockDim.x + threadIdx.x;
    if (n >= NPTS) return;

    float fy = locs[2*n + 0];
    float fx = locs[2*n + 1];
    int yi = (int)fy; yi = yi < 0 ? 0 : (yi > IMG-1 ? IMG-1 : yi);
    int xi = (int)fx; xi = xi < 0 ? 0 : (xi > IMG-1 ? IMG-1 : xi);

    const float* b0 = batch + (size_t)imgid[n] * 3 * IMG * IMG;
    auto gray = [&](int y, int x) -> float {
        if (y < 0 || y >= IMG || x < 0 || x >= IMG) return 0.0f;
        int o = y * IMG + x;
        return (b0[o] + b0[IMG*IMG + o] + b0[2*IMG*IMG + o]) * (1.0f / 3.0f);
    };

    float g00 = gray(yi-1, xi-1), g01 = gray(yi-1, xi), g02 = gray(yi-1, xi+1);
    float g10 = gray(yi,   xi-1),                       g12 = gray(yi,   xi+1);
    float g20 = gray(yi+1, xi-1), g21 = gray(yi+1, xi), g22 = gray(yi+1, xi+1);

    float gx = (g02 - g00) + 2.0f * (g12 - g10) + (g22 - g20);
    float gy = (g20 - g00) + 2.0f * (g21 - g01) + (g22 - g02);

    float inv = 1.0f / (sqrtf(gx*gx + gy*gy) + 1e-8f);
    norms[2*n + 0] = gx * inv;
    norms[2*n + 1] = gy * inv;
}

// sorted-ascending 8-entry insertion (call only when d < kd[KNN-1])
__device__ __forceinline__ void insert8(float (&kd)[KNN], int (&kj)[KNN],
                                        float d, int j)
{
    kd[KNN-1] = d; kj[KNN-1] = j;
#pragma unroll
    for (int s = KNN-1; s >= 1; --s) {
        if (kd[s] < kd[s-1]) {
            float td = kd[s]; kd[s] = kd[s-1]; kd[s-1] = td;
            int   tj = kj[s]; kj[s] = kj[s-1]; kj[s-1] = tj;
        }
    }
}

// ---------------------------------------------------------------------------
// Kernel 2: top-8 nearest same-image neighbors per point.
// Block = 8 waves over one 32-row group. Wave w scans columns
// [w*1024, (w+1)*1024) with v_wmma_f32_16x16x4_f32 producing exact d2 tiles
// (A row = (x, y, |p|^2, 1), B col = (-2x, -2y, 1, |p|^2)). Partial top-8
// lists are merged per row through LDS after a block barrier.
// ---------------------------------------------------------------------------
__global__ __launch_bounds__(COL_WAVES * 32)
void knn_kernel(const float* __restrict__ locs,
                const int*   __restrict__ imgid,
                float*       __restrict__ knn_d2,
                int*         __restrict__ knn_idx)
{
    __shared__ __align__(16) float tile[COL_WAVES][ROWS_PER_WAVE][16]; // 16 KB
    __shared__ float part_d[COL_WAVES][ROWS_PER_WAVE][KNN];            //  8 KB
    __shared__ int   part_j[COL_WAVES][ROWS_PER_WAVE][KNN];            //  8 KB

    const int lane = threadIdx.x & 31;
    const int wave = threadIdx.x >> 5;
    const int i0   = blockIdx.x * ROWS_PER_WAVE;
    const int half = (lane < 16) ? 0 : 1;   // which K-pair this lane holds
    const int off  = half * 8;              // C-layout row offset for upper half
    const int m    = lane & 15;

    // A matrices (f32 16x4 layout: lanes 0-15 -> K=0,1 ; lanes 16-31 -> K=2,3)
    v2f A[2];
#pragma unroll
    for (int t = 0; t < 2; ++t) {
        int i = i0 + 16*t + m;
        float x = locs[2*i + 0], y = locs[2*i + 1];
        v2f a;
        if (half) { a.x = x*x + y*y; a.y = 1.0f; }
        else      { a.x = x;         a.y = y;    }
        A[t] = a;
    }

    // imgid of the 16 D-rows this lane's accumulator VGPRs correspond to
    int rimg[16];
#pragma unroll
    for (int t = 0; t < 2; ++t)
#pragma unroll
        for (int v = 0; v < 8; ++v)
            rimg[t*8 + v] = imgid[i0 + 16*t + v + off];

    float kd[KNN]; int kj[KNN];
#pragma unroll
    for (int q = 0; q < KNN; ++q) { kd[q] = __builtin_inff(); kj[q] = 0; }

    const float* row_lds = &tile[wave][lane][0];
    const v8f cz = {0.f, 0.f, 0.f, 0.f, 0.f, 0.f, 0.f, 0.f};
    const int jbeg = wave * COLS_PER_WAVE;
    const int jend = jbeg + COLS_PER_WAVE;

    for (int j0 = jbeg; j0 < jend; j0 += 16) {
        int jp = j0 + 256;
        if (jp < jend) __builtin_prefetch(&locs[2*jp + 2*m], 0, 1);

        int j = j0 + m;
        float xj = locs[2*j + 0], yj = locs[2*j + 1];
        v2f B;
        if (half) { B.x = 1.0f;       B.y = xj*xj + yj*yj; }
        else      { B.x = -2.0f * xj; B.y = -2.0f * yj;    }
        int cimg = imgid[j];

        v8f acc0 = __builtin_amdgcn_wmma_f32_16x16x4_f32(
            false, A[0], false, B, (short)0, cz, false, false);
        v8f acc1 = __builtin_amdgcn_wmma_f32_16x16x4_f32(
            false, A[1], false, B, (short)0, cz, false, false);

        // self-pairs only possible when this column tile overlaps our rows
        bool diag = (j0 < i0 + ROWS_PER_WAVE) && (j0 + 16 > i0);  // uniform
        if (diag) {
#pragma unroll
            for (int v = 0; v < 8; ++v) {
                int r0 = i0 + v + off, r1 = i0 + 16 + v + off;
                float d0 = ((rimg[v]   != cimg) || (r0 == j)) ? __builtin_inff() : acc0[v];
                float d1 = ((rimg[8+v] != cimg) || (r1 == j)) ? __builtin_inff() : acc1[v];
                tile[wave][v + off][m]      = d0;
                tile[wave][16 + v + off][m] = d1;
            }
        } else {
#pragma unroll
            for (int v = 0; v < 8; ++v) {
                float d0 = (rimg[v]   != cimg) ? __builtin_inff() : acc0[v];
                float d1 = (rimg[8+v] != cimg) ? __builtin_inff() : acc1[v];
                tile[wave][v + off][m]      = d0;
                tile[wave][16 + v + off][m] = d1;
            }
        }
        __builtin_amdgcn_wave_barrier();   // same-wave DS is in-order; fence compiler

        // lane scans its own row (global row i0+lane) of the 32x16 tile
        const float4* rv = (const float4*)row_lds;
        float4 a0 = rv[0], a1 = rv[1], a2 = rv[2], a3 = rv[3];
        float mn = fminf(fminf(fminf(fminf(a0.x, a0.y), fminf(a0.z, a0.w)),
                               fminf(fminf(a1.x, a1.y), fminf(a1.z, a1.w))),
                         fminf(fminf(fminf(a2.x, a2.y), fminf(a2.z, a2.w)),
                               fminf(fminf(a3.x, a3.y), fminf(a3.z, a3.w))));
        if (mn < kd[KNN-1]) {              // rare path: something worth inserting
            float dv[16] = { a0.x, a0.y, a0.z, a0.w, a1.x, a1.y, a1.z, a1.w,
                             a2.x, a2.y, a2.z, a2.w, a3.x, a3.y, a3.z, a3.w };
#pragma unroll
            for (int e = 0; e < 16; ++e) {
                float d = dv[e];
                if (d < kd[KNN-1]) insert8(kd, kj, d, j0 + e);
            }
        }
        __builtin_amdgcn_wave_barrier();   // WAR fence before next tile's stores
    }

    // publish this wave's partial top-8 for its 32 rows
#pragma unroll
    for (int q = 0; q < KNN; ++q) {
        part_d[wave][lane][q] = kd[q];
        part_j[wave][lane][q] = kj[q];
    }
    __syncthreads();

    // first wave merges the 8 partial lists per row (each sorted ascending)
    if (threadIdx.x < ROWS_PER_WAVE) {
        int r = threadIdx.x;
        float md[KNN]; int mj[KNN];
#pragma unroll
        for (int q = 0; q < KNN; ++q) { md[q] = __builtin_inff(); mj[q] = 0; }
#pragma unroll
        for (int w = 0; w < COL_WAVES; ++w) {
#pragma unroll
            for (int q = 0; q < KNN; ++q) {
                float d = part_d[w][r][q];
                if (d >= md[KNN-1]) break;          // list sorted: rest can't win
                insert8(md, mj, d, part_j[w][r][q]);
            }
        }
        int i = i0 + r;
#pragma unroll
        for (int q = 0; q < KNN; ++q) {
            knn_d2 [i*KNN + q] = md[q];
            knn_idx[i*KNN + q] = mj[q];
        }
    }
}

// ---------------------------------------------------------------------------
// Kernel 3: emit strs (N*K rows) and new_strs (N*K*K rows), 7 floats each.
// ---------------------------------------------------------------------------
__global__ __launch_bounds__(256)
void emit_kernel(const float* __restrict__ locs,
                 const float* __restrict__ norms,
                 const float* __restrict__ knn_d2,
                 const int*   __restrict__ knn_idx,
                 float*       __restrict__ out)
{
    int t = blockIdx.x * blockDim.x + threadIdx.x;
    if (t >= NPTS * KNN) return;
    int i = t / KNN;

    float lix = locs[2*i],  liy = locs[2*i + 1];
    float nix = norms[2*i], niy = norms[2*i + 1];

    int   j   = knn_idx[t];
    float d2  = knn_d2[t];
    float em  = (d2 <= RAD2) ? 1.0f : 0.0f;
    float fjx = locs[2*j], fjy = locs[2*j + 1];
    float snx = 0.5f * (nix + norms[2*j]);
    float sny = 0.5f * (niy + norms[2*j + 1]);

    float* o = out + (size_t)t * 7;
    o[0] = lix * em; o[1] = liy * em;
    o[2] = fjx * em; o[3] = fjy * em;
    o[4] = snx * em; o[5] = sny * em;
    o[6] = 0.0f;

    const size_t base2 = (size_t)NPTS * KNN * 7;
    const int ib = i * KNN;
#pragma unroll
    for (int l = 0; l < KNN; ++l) {
        int   j2   = knn_idx[ib + l];
        float d2l  = knn_d2[ib + l];
        bool  eml  = (d2l <= RAD2);
        float frx  = locs[2*j2], fry = locs[2*j2 + 1];
        float dx   = fjx - frx,  dy  = fjy - fry;
        float seg  = sqrtf(dx*dx + dy*dy);
        bool  valid = (em != 0.0f) && eml && (seg > 0.001f);
        float seg_s = valid ? seg : 1.0f;
        float dev  = (fry - fjy) * (fjx - lix) - (fjy - liy) * (frx - fjx);
        float frac = dev / (seg_s * seg_s);
        float snlx = 0.5f * (nix + norms[2*j2]);
        float snly = 0.5f * (niy + norms[2*j2 + 1]);
        float nnx  = 0.5f * (snx + snlx);
        float nny  = 0.5f * (sny + snly);
        float innr = 1.0f / (sqrtf(nnx*nnx + nny*nny) + 1e-8f);
        nnx *= innr; nny *= innr;
        float vm = valid ? 1.0f : 0.0f;

        float* o2 = out + base2 + ((size_t)t * KNN + l) * 7;
        o2[0] = fjx * vm; o2[1] = fjy * vm;
        o2[2] = frx * vm; o2[3] = fry * vm;
        o2[4] = nnx * vm; o2[5] = nny * vm;
        o2[6] = frac * vm;
    }
}

// ---------------------------------------------------------------------------
extern "C" void kernel_launch(void* const* d_in, const int* in_sizes, int n_in,
                              void* d_out, int out_size, void* d_ws, size_t ws_size,
                              hipStream_t stream)
{
    const float* batch = (const float*)d_in[0];   // (8,3,256,256) f32
    const float* locs  = (const float*)d_in[1];   // (8192,2) f32
    const int*   imgid = (const int*)  d_in[2];   // (8192,) i32
    float* out = (float*)d_out;

    // workspace layout: norms (2N f32) | knn_d2 (N*K f32) | knn_idx (N*K i32)
    float* norms   = (float*)d_ws;
    float* knn_d2  = norms + 2 * NPTS;
    int*   knn_idx = (int*)(knn_d2 + NPTS * KNN);

    sobel_norms_kernel<<<NPTS / 256, 256, 0, stream>>>(batch, locs, imgid, norms);
    knn_kernel<<<NPTS / ROWS_PER_WAVE, COL_WAVES * 32, 0, stream>>>(
        locs, imgid, knn_d2, knn_idx);
    emit_kernel<<<(NPTS * KNN) / 256, 256, 0, stream>>>(locs, norms, knn_d2, knn_idx, out);
}